// AttentionFusion_40785009443425
// MI455X (gfx1250) — compile-verified
//
#include <hip/hip_runtime.h>
#include <math.h>

typedef __attribute__((ext_vector_type(2))) float v2f;
typedef __attribute__((ext_vector_type(8))) float v8f;
typedef __attribute__((ext_vector_type(4))) int   v4i;

#define BATCH 8192
#define NSRC  6
#define HDIM  1024
#define H2DIM 2048

// ---------------------------------------------------------------------------
// CDNA5 async global->LDS staging (ASYNCcnt path), guarded so fallback always
// compiles. Builtin signature (probe-confirmed via diagnostic):
//   (v4i __device__*, v4i __shared__*, imm int offset, imm int cpol)
// ---------------------------------------------------------------------------
#if defined(__has_builtin)
#  if __has_builtin(__builtin_amdgcn_global_load_async_to_lds_b128)
#    define USE_ASYNC_LDS 1
#  endif
#  if __has_builtin(__builtin_amdgcn_s_wait_asynccnt)
#    define HAVE_WAIT_ASYNC_BUILTIN 1
#  endif
#endif

__device__ __forceinline__ void stage_b128(const float* __restrict__ g,
                                           float* __restrict__ l) {
#ifdef USE_ASYNC_LDS
    __builtin_amdgcn_global_load_async_to_lds_b128(
        (__attribute__((address_space(1))) v4i*)(size_t)g,
        (__attribute__((address_space(3))) v4i*)l,
        0, 0);
#else
    float4 v = *(const float4*)g;
    *(float4*)l = v;
#endif
}

__device__ __forceinline__ void wait_async_lds() {
#ifdef USE_ASYNC_LDS
#  ifdef HAVE_WAIT_ASYNC_BUILTIN
    __builtin_amdgcn_s_wait_asynccnt(0);
#  else
    asm volatile("s_wait_asynccnt 0" ::: "memory");
#  endif
#endif
}

// ---------------------------------------------------------------------------
// 32x32 tiled transpose: in [R,C] -> out [C,R]
// ---------------------------------------------------------------------------
template<int R, int C>
__global__ __launch_bounds__(256)
void transpose_kernel(const float* __restrict__ in, float* __restrict__ out) {
    __shared__ float tile[32][33];
    const int bx = blockIdx.x * 32;   // col base in input
    const int by = blockIdx.y * 32;   // row base in input
    const int tx = threadIdx.x & 31;
    const int ty = threadIdx.x >> 5;  // 8 rows per pass
#pragma unroll
    for (int i = 0; i < 32; i += 8)
        tile[ty + i][tx] = in[(size_t)(by + ty + i) * C + bx + tx];
    __syncthreads();
#pragma unroll
    for (int i = 0; i < 32; i += 8)
        out[(size_t)(bx + ty + i) * R + by + tx] = tile[tx][ty + i];
}

// ---------------------------------------------------------------------------
// Kernel 1: mean/max pooling over S -> cond[b,0:H]=mean, cond[b,H:2H]=max
// ---------------------------------------------------------------------------
__global__ __launch_bounds__(256)
void pool_kernel(const float* __restrict__ E, float* __restrict__ cond) {
    int idx = blockIdx.x * blockDim.x + threadIdx.x;
    if (idx >= BATCH * HDIM) return;
    int b = idx >> 10;
    int h = idx & (HDIM - 1);
    const float* e = E + (size_t)b * NSRC * HDIM + h;
    float mx = -INFINITY, sm = 0.f;
#pragma unroll
    for (int s = 0; s < NSRC; ++s) {
        float v = e[(size_t)s * HDIM];
        sm += v;
        mx = fmaxf(mx, v);
    }
    cond[(size_t)b * H2DIM + h]        = sm * (1.f / (float)NSRC);
    cond[(size_t)b * H2DIM + HDIM + h] = mx;
}

// ---------------------------------------------------------------------------
// WMMA f32 GEMM: C[M,1024] = A[M,K] * B[n][k]  (B given [N,K] row-major)
//   K, LDA compile-time -> shift/add addressing.
//   EPI=0: none. EPI=1: += bias0[n]+bias1[n]+gate[m]*wp[n]. EPI=2: += bias0[n]
// BM=128, BN=128, BK=16; 8 waves; each wave 32x64 (2x4 WMMA frags).
// Double-buffered LDS, async global->LDS staging, ONE barrier per K-tile:
// refills of a buffer only occur after the next top barrier, which every wave
// reaches only after finishing its reads of that buffer.
// ---------------------------------------------------------------------------
template<int K, int LDA, int EPI>
__global__ __launch_bounds__(256)
void gemm_wmma_f32(const float* __restrict__ A, const float* __restrict__ Bm,
                   float* __restrict__ C,
                   const float* __restrict__ bias0, const float* __restrict__ bias1,
                   const float* __restrict__ wp, const float* __restrict__ gate) {
    constexpr int BM = 128, BN = 128, BK = 16;
    constexpr int PAD = 20;                 // 80B rows: 16B-aligned + conflict-free
    constexpr int N = HDIM;
    constexpr int ABUF = BM * PAD, BBUF = BN * PAD;
    __shared__ float sA[2 * ABUF];          // 40960 B total LDS
    __shared__ float sB[2 * BBUF];

    const int tid  = threadIdx.x;
    const int wave = tid >> 5;
    const int lane = tid & 31;
    const int m0 = blockIdx.y * BM;
    const int n0 = blockIdx.x * BN;
    const int wm = (wave & 3) * 32;         // 4 waves tile M
    const int wn = (wave >> 2) * 64;        // 2 waves tile N

    // staging geometry: 2048 floats per tile, 8 floats/thread (2x b128)
    const int tr = tid >> 1;
    const int tc = (tid & 1) * 8;
    const float* gA = A  + (size_t)(m0 + tr) * LDA + tc;
    const float* gB = Bm + (size_t)(n0 + tr) * K + tc;
    float* lA = &sA[tr * PAD + tc];
    float* lB = &sB[tr * PAD + tc];

    v8f acc[2][4] = {};
    const int lrow = lane & 15;
    const int krow = (lane >> 4) * 2;       // f32 WMMA K sub-offset per lane half

    // prologue: stage tile 0 into buffer 0
    stage_b128(gA,     lA);
    stage_b128(gA + 4, lA + 4);
    stage_b128(gB,     lB);
    stage_b128(gB + 4, lB + 4);

    constexpr int NT = K / BK;
    for (int t = 0; t < NT; ++t) {
        const int cur = t & 1;
        wait_async_lds();                   // my async fills of `cur` complete
        __syncthreads();                    // => everyone's fills of `cur` visible;
                                            //    also: everyone done reading `nxt`
        if (t + 1 < NT) {
            const int nxt = cur ^ 1;
            const int k0  = (t + 1) * BK;
            stage_b128(gA + k0,     lA + nxt * ABUF);
            stage_b128(gA + k0 + 4, lA + nxt * ABUF + 4);
            stage_b128(gB + k0,     lB + nxt * BBUF);
            stage_b128(gB + k0 + 4, lB + nxt * BBUF + 4);
        }
        const float* bufA = &sA[cur * ABUF + (wm + lrow) * PAD + krow];
        const float* bufB = &sB[cur * BBUF + (wn + lrow) * PAD + krow];
#pragma unroll
        for (int kk = 0; kk < BK; kk += 4) {
            v2f a[2], b[4];
#pragma unroll
            for (int i = 0; i < 2; ++i) {
                a[i].x = bufA[i * 16 * PAD + kk];
                a[i].y = bufA[i * 16 * PAD + kk + 1];
            }
#pragma unroll
            for (int j = 0; j < 4; ++j) {
                b[j].x = bufB[j * 16 * PAD + kk];
                b[j].y = bufB[j * 16 * PAD + kk + 1];
            }
#pragma unroll
            for (int i = 0; i < 2; ++i)
#pragma unroll
                for (int j = 0; j < 4; ++j)
                    acc[i][j] = __builtin_amdgcn_wmma_f32_16x16x4_f32(
                        false, a[i], false, b[j], (short)0, acc[i][j], false, false);
        }
        // no trailing barrier: next iteration's wait+barrier orders buffer reuse
    }

    // epilogue: D layout = VGPR r -> M=r (lanes<16) / M=r+8 (lanes>=16)
    const int lhi = lane >> 4;
#pragma unroll
    for (int i = 0; i < 2; ++i) {
#pragma unroll
        for (int j = 0; j < 4; ++j) {
            int col = n0 + wn + j * 16 + lrow;
            float cb = 0.f;
            if (EPI == 1) cb = bias0[col] + bias1[col];
            if (EPI == 2) cb = bias0[col];
#pragma unroll
            for (int r = 0; r < 8; ++r) {
                int row = m0 + wm + i * 16 + r + 8 * lhi;
                float v = acc[i][j][r] + cb;
                if (EPI == 1) v += gate[row] * wp[col];
                C[(size_t)row * N + col] = v;
            }
        }
    }
}

// ---------------------------------------------------------------------------
// Kernel 4: one wave32 per batch row b.
//   scores[s] = E[b,s]·kq[b] + bk·q[b] + source_bias[s]; softmax; entropy;
//   pooled[b] = sum_s w_s * E[b,s]
// ---------------------------------------------------------------------------
__global__ __launch_bounds__(256)
void attn_softmax_pool(const float* __restrict__ E, const float* __restrict__ kq,
                       const float* __restrict__ q,  const float* __restrict__ bk,
                       const float* __restrict__ sbias,
                       float* __restrict__ pooled, float* __restrict__ wOut,
                       float* __restrict__ entOut) {
    const int lane = threadIdx.x & 31;
    const int b = blockIdx.x * (blockDim.x >> 5) + (threadIdx.x >> 5);
    if (b >= BATCH) return;

    const float* e   = E  + (size_t)b * NSRC * HDIM;
    const float* kqb = kq + (size_t)b * HDIM;
    const float* qb  = q  + (size_t)b * HDIM;

    float rk[HDIM / 32];
    float pb = 0.f;
#pragma unroll
    for (int i = 0; i < HDIM / 32; ++i) {
        int h = i * 32 + lane;
        rk[i] = kqb[h];
        pb += bk[h] * qb[h];
    }

    float sc[NSRC];
#pragma unroll
    for (int s = 0; s < NSRC; ++s) {
        float p = 0.f;
#pragma unroll
        for (int i = 0; i < HDIM / 32; ++i)
            p += e[(size_t)s * HDIM + i * 32 + lane] * rk[i];
        sc[s] = p;
    }
#pragma unroll
    for (int s = 0; s < NSRC; ++s) {
        float v = sc[s];
#pragma unroll
        for (int o = 16; o > 0; o >>= 1) v += __shfl_xor(v, o, 32);
        sc[s] = v;
    }
#pragma unroll
    for (int o = 16; o > 0; o >>= 1) pb += __shfl_xor(pb, o, 32);

    float mx = -INFINITY;
#pragma unroll
    for (int s = 0; s < NSRC; ++s) { sc[s] += pb + sbias[s]; mx = fmaxf(mx, sc[s]); }
    float w[NSRC], sum = 0.f;
#pragma unroll
    for (int s = 0; s < NSRC; ++s) { w[s] = expf(sc[s] - mx); sum += w[s]; }
    float inv = 1.f / sum;
    float ent = 0.f;
#pragma unroll
    for (int s = 0; s < NSRC; ++s) { w[s] *= inv; ent -= w[s] * logf(w[s] + 1e-8f); }

#pragma unroll
    for (int i = 0; i < HDIM / 32; ++i) {
        int h = i * 32 + lane;
        float p = 0.f;
#pragma unroll
        for (int s = 0; s < NSRC; ++s) p += w[s] * e[(size_t)s * HDIM + h];
        pooled[(size_t)b * HDIM + h] = p;
    }

    if (lane == 0) {
#pragma unroll
        for (int s = 0; s < NSRC; ++s) wOut[(size_t)b * NSRC + s] = w[s];
        entOut[b] = ent;
    }
}

// ---------------------------------------------------------------------------
// Launch: transposes -> pooling -> query GEMM -> kq GEMM -> softmax/pool ->
//         value GEMM
// ---------------------------------------------------------------------------
extern "C" void kernel_launch(void* const* d_in, const int* in_sizes, int n_in,
                              void* d_out, int out_size, void* d_ws, size_t ws_size,
                              hipStream_t stream) {
    const float* E     = (const float*)d_in[0];   // [B,S,H]
    const float* gate  = (const float*)d_in[2];   // [B,1]
    const float* Wk    = (const float*)d_in[3];   // [H,H]  (used as [N,K] directly)
    const float* bk    = (const float*)d_in[4];   // [H]
    const float* Wv    = (const float*)d_in[5];   // [H,H]
    const float* bv    = (const float*)d_in[6];   // [H]
    const float* Wq    = (const float*)d_in[7];   // [2H,H]
    const float* bq    = (const float*)d_in[8];   // [H]
    const float* Wp    = (const float*)d_in[9];   // [1,H]
    const float* bp    = (const float*)d_in[10];  // [H]
    const float* sbias = (const float*)d_in[11];  // [S]

    float* fusedOut = (float*)d_out;                       // [B,H]
    float* wOut     = fusedOut + (size_t)BATCH * HDIM;     // [B,S]
    float* entOut   = wOut + (size_t)BATCH * NSRC;         // [B]

    float* ws     = (float*)d_ws;
    float* cond   = ws;                                    // [B,2H]
    float* query  = ws + (size_t)BATCH * H2DIM;            // [B,H]
    float* kqbuf  = ws;                                    // [B,H] (aliases cond)
    float* pooled = ws + (size_t)BATCH * HDIM;             // [B,H] (2nd half of cond)
    float* WqT    = ws + (size_t)BATCH * H2DIM + (size_t)BATCH * HDIM;  // [H,2H]
    float* WvT    = WqT + (size_t)HDIM * H2DIM;                         // [H,H]

    // 0) pre-transpose weights so all GEMM B-operands are [N,K] row-major
    transpose_kernel<H2DIM, HDIM><<<dim3(HDIM / 32, H2DIM / 32), 256, 0, stream>>>(Wq, WqT);
    transpose_kernel<HDIM,  HDIM><<<dim3(HDIM / 32, HDIM  / 32), 256, 0, stream>>>(Wv, WvT);

    // 1) pooling -> cond
    pool_kernel<<<(BATCH * HDIM + 255) / 256, 256, 0, stream>>>(E, cond);

    // 2) query = cond @ Wq + bq + bp + gate*Wp      [B,2048]x[2048,1024]
    gemm_wmma_f32<H2DIM, H2DIM, 1><<<dim3(HDIM / 128, BATCH / 128), 256, 0, stream>>>(
        cond, WqT, query, bq, bp, Wp, gate);

    // 3) kq = query @ Wk^T                          [B,1024]x[1024,1024]
    gemm_wmma_f32<HDIM, HDIM, 0><<<dim3(HDIM / 128, BATCH / 128), 256, 0, stream>>>(
        query, Wk, kqbuf, nullptr, nullptr, nullptr, nullptr);

    // 4) scores/softmax/entropy + weighted pooling (one wave32 per row)
    attn_softmax_pool<<<BATCH / 8, 256, 0, stream>>>(E, kqbuf, query, bk, sbias,
                                                     pooled, wOut, entOut);

    // 5) fused = pooled @ Wv + bv                   [B,1024]x[1024,1024]
    gemm_wmma_f32<HDIM, HDIM, 2><<<dim3(HDIM / 128, BATCH / 128), 256, 0, stream>>>(
        pooled, WvT, fusedOut, bv, nullptr, nullptr, nullptr);
}